// DualGraphConv_87866440942275
// MI455X (gfx1250) — compile-verified
//
#include <hip/hip_runtime.h>
#include <hip/hip_bf16.h>

typedef float v2f __attribute__((ext_vector_type(2)));
typedef float v8f __attribute__((ext_vector_type(8)));

#define DIM 132   // padded LDS stride (bank-conflict-free, 16B aligned)

// ---------------------------------------------------------------------------
// COO SpMM scatter: agg[rows[e]] += vals[e] * x[cols[e]]   (one wave per edge)
// ---------------------------------------------------------------------------
__global__ __launch_bounds__(256) void spmm_kernel(
    const int* __restrict__ rows, const int* __restrict__ cols,
    const float* __restrict__ vals, const float* __restrict__ x,
    float* __restrict__ out, int n_edges)
{
    int e = blockIdx.x * 8 + (threadIdx.x >> 5);
    if (e >= n_edges) return;
    int lane = threadIdx.x & 31;
    int r = rows[e];
    int c = cols[e];
    float v = vals[e];
    const float4 xv = *(const float4*)(x + (size_t)c * 128 + lane * 4);
    float* o = out + (size_t)r * 128 + lane * 4;
    __hip_atomic_fetch_add(o + 0, v * xv.x, __ATOMIC_RELAXED, __HIP_MEMORY_SCOPE_AGENT);
    __hip_atomic_fetch_add(o + 1, v * xv.y, __ATOMIC_RELAXED, __HIP_MEMORY_SCOPE_AGENT);
    __hip_atomic_fetch_add(o + 2, v * xv.z, __ATOMIC_RELAXED, __HIP_MEMORY_SCOPE_AGENT);
    __hip_atomic_fetch_add(o + 3, v * xv.w, __ATOMIC_RELAXED, __HIP_MEMORY_SCOPE_AGENT);
}

// ---------------------------------------------------------------------------
// Fused: dst = LayerNorm( cur + leaky_relu(agg @ W^T + bias) ) * gamma + beta
// Wave32 WMMA f32 16x16x4. One wave -> 16 rows x 128 cols.
// ---------------------------------------------------------------------------
__global__ __launch_bounds__(256) void gemm_ln_kernel(
    const float* __restrict__ agg,
    const float* __restrict__ Wm,     // [128,128] row-major; out = agg @ Wm^T
    const float* __restrict__ bias,   // [128]
    const float* __restrict__ gam,    // [128]
    const float* __restrict__ bet,    // [128]
    const float* __restrict__ cur,    // [n,128] residual input
    float* __restrict__ dst,          // [n,128]
    int n_rows)
{
    extern __shared__ float smem[];
    float* lW = smem;                 // 128 * DIM floats
    float* lA = smem + 128 * DIM;     // 8 waves * 16 * DIM floats

    const int tid  = threadIdx.x;
    const int wave = tid >> 5;
    const int lane = tid & 31;
    const int half = lane >> 4;       // 0: lanes 0-15, 1: lanes 16-31
    const int l15  = lane & 15;
    const int koff = half * 2;        // per ISA A/B fragment layout (K split)

    // ---- stage W into LDS (each thread: one half-row, 16x float4) ----
    {
        int r  = tid >> 1;
        int ch = (tid & 1) * 64;
        const float4* src = (const float4*)(Wm + r * 128 + ch);
        #pragma unroll
        for (int j = 0; j < 16; ++j) {
            float4 v = src[j];
            *(float4*)(lW + r * DIM + ch + 4 * j) = v;
        }
    }

    // ---- stage this wave's 16x128 A tile (row-clamped, coalesced) ----
    const int row0 = blockIdx.x * 128 + wave * 16;
    float* myA = lA + wave * 16 * DIM;
    #pragma unroll
    for (int i = 0; i < 16; ++i) {
        int rr = row0 + i;
        if (rr > n_rows - 1) rr = n_rows - 1;
        float4 v = *(const float4*)(agg + (size_t)rr * 128 + 4 * lane);
        *(float4*)(myA + i * DIM + 4 * lane) = v;
    }
    __syncthreads();

    // ---- K loop: D = A x B + C with V_WMMA_F32_16X16X4_F32 ----
    v8f acc[8] = {};
    for (int k = 0; k < 128; k += 4) {
        // A frag: lane(l<16): A[l][k], A[l][k+1]; lane(l>=16): A[l-16][k+2..3]
        v2f a = *(const v2f*)(myA + l15 * DIM + k + koff);
        #pragma unroll
        for (int t = 0; t < 8; ++t) {
            // B frag: B[kk][n] = W[n][kk], n = t*16 + l15
            v2f b = *(const v2f*)(lW + (t * 16 + l15) * DIM + k + koff);
            acc[t] = __builtin_amdgcn_wmma_f32_16x16x4_f32(
                false, a, false, b, (short)0, acc[t], false, false);
        }
    }

    // ---- per-lane column params ----
    float bia[8], gg[8], bb[8];
    #pragma unroll
    for (int t = 0; t < 8; ++t) {
        int c = t * 16 + l15;
        bia[t] = bias[c]; gg[t] = gam[c]; bb[t] = bet[c];
    }

    // C layout: acc[t][v] = out[row0 + v + 8*half][t*16 + l15]
    #pragma unroll
    for (int v = 0; v < 8; ++v) {
        const int row = row0 + v + half * 8;
        const int rc  = row < n_rows ? row : n_rows - 1;
        float x[8];
        float s = 0.f, ss = 0.f;
        #pragma unroll
        for (int t = 0; t < 8; ++t) {
            float val = acc[t][v] + bia[t];
            val = val > 0.f ? val : 0.01f * val;               // leaky_relu
            val += cur[(size_t)rc * 128 + t * 16 + l15];       // residual
            x[t] = val; s += val; ss += val * val;
        }
        // row statistics: butterfly over the 16 lanes holding this row
        #pragma unroll
        for (int m = 1; m < 16; m <<= 1) {
            s  += __shfl_xor(s,  m, 32);
            ss += __shfl_xor(ss, m, 32);
        }
        float mu  = s * (1.0f / 128.0f);
        float var = ss * (1.0f / 128.0f) - mu * mu;
        float rs  = rsqrtf(var + 1e-5f);
        if (row < n_rows) {
            #pragma unroll
            for (int t = 0; t < 8; ++t)
                dst[(size_t)row * 128 + t * 16 + l15] = (x[t] - mu) * rs * gg[t] + bb[t];
        }
    }
}

// ---------------------------------------------------------------------------
extern "C" void kernel_launch(void* const* d_in, const int* in_sizes, int n_in,
                              void* d_out, int out_size, void* d_ws, size_t ws_size,
                              hipStream_t stream)
{
    const float* user_emb = (const float*)d_in[0];
    const float* item_emb = (const float*)d_in[1];
    const float* ui_vals  = (const float*)d_in[2];
    const float* so_vals  = (const float*)d_in[3];
    const float* ui_W  = (const float*)d_in[4];
    const float* ui_b  = (const float*)d_in[5];
    const float* ui_g  = (const float*)d_in[6];
    const float* ui_lb = (const float*)d_in[7];
    const float* so_W  = (const float*)d_in[8];
    const float* so_b  = (const float*)d_in[9];
    const float* so_g  = (const float*)d_in[10];
    const float* so_lb = (const float*)d_in[11];
    const int* ui_rows = (const int*)d_in[12];
    const int* ui_cols = (const int*)d_in[13];
    const int* so_rows = (const int*)d_in[14];
    const int* so_cols = (const int*)d_in[15];

    const int Dm  = 128;
    const int U   = in_sizes[0] / Dm;
    const int I   = in_sizes[1] / Dm;
    const int Eui = in_sizes[2];
    const int Eso = in_sizes[3];
    const int L   = in_sizes[4] / (Dm * Dm);

    float* out = (float*)d_out;
    size_t slab = (size_t)U * Dm;
    float* ui_stack = out;                              // [L+1, U, D]
    float* so_stack = out + (size_t)(L + 1) * slab;     // [L+1, U, D]
    float* ci_out   = so_stack + (size_t)(L + 1) * slab;// [I, D]

    float* agg = (float*)d_ws;                          // max(U,I)*D f32 scratch

    // layer-0 slots = user_emb
    hipMemcpyAsync(ui_stack, user_emb, slab * sizeof(float), hipMemcpyDeviceToDevice, stream);
    hipMemcpyAsync(so_stack, user_emb, slab * sizeof(float), hipMemcpyDeviceToDevice, stream);

    const dim3 blk(256);
    const size_t shmem = (size_t)(128 * DIM + 8 * 16 * DIM) * sizeof(float);

    auto run_spmm = [&](const int* r, const int* c, const float* v,
                        const float* x, int e, int nout) {
        hipMemsetAsync(agg, 0, (size_t)nout * Dm * sizeof(float), stream);
        spmm_kernel<<<dim3((e + 7) / 8), blk, 0, stream>>>(r, c, v, x, agg, e);
    };
    auto run_gemm = [&](const float* Wm, const float* bi, const float* g,
                        const float* be, const float* cu, float* ds, int n) {
        gemm_ln_kernel<<<dim3((n + 127) / 128), blk, shmem, stream>>>(
            agg, Wm, bi, g, be, cu, ds, n);
    };

    // ---- user-item bipartite conv ----
    const float* ci_old = item_emb;
    for (int i = 0; i < L; ++i) {
        const float* cu_old = ui_stack + (size_t)i * slab;
        float* cu_new = ui_stack + (size_t)(i + 1) * slab;
        // ua: aggregate items into users, update users
        run_spmm(ui_rows, ui_cols, ui_vals, ci_old, Eui, U);
        run_gemm(ui_W + (size_t)i * Dm * Dm, ui_b + i * Dm, ui_g + i * Dm,
                 ui_lb + i * Dm, cu_old, cu_new, U);
        // ia: aggregate (old) users into items, update items
        run_spmm(ui_cols, ui_rows, ui_vals, cu_old, Eui, I);
        run_gemm(ui_W + (size_t)i * Dm * Dm, ui_b + i * Dm, ui_g + i * Dm,
                 ui_lb + i * Dm, ci_old, ci_out, I);
        ci_old = ci_out;   // layer >=1 runs in-place on ci_out (per-thread safe)
    }

    // ---- social conv ----
    for (int i = 0; i < L; ++i) {
        const float* cs_old = so_stack + (size_t)i * slab;
        float* cs_new = so_stack + (size_t)(i + 1) * slab;
        run_spmm(so_rows, so_cols, so_vals, cs_old, Eso, U);
        run_gemm(so_W + (size_t)i * Dm * Dm, so_b + i * Dm, so_g + i * Dm,
                 so_lb + i * Dm, cs_old, cs_new, U);
    }

    (void)n_in; (void)out_size; (void)ws_size;
}